// ROIPooling3D_24481313587460
// MI455X (gfx1250) — compile-verified
//
#include <hip/hip_runtime.h>
#include <stdint.h>

// ROI adaptive max-pool 3D for MI455X (gfx1250).
// Memory-bound gather: reads exactly Lz*Ly*Lx per (roi,channel) (~2.4 GB total)
// vs ~600M fmax ops -> bound by HBM/L2 bandwidth (23.3 TB/s floor ~100us;
// 192MB L2 captures most cross-ROI reuse of the 128MB feature tensor).
// Strategy: Tensor Data Mover DMAs each (b,c,z) ROI slice (strided 2D tile of
// the [B,C,D,H,W] tensor) into LDS, triple-buffered with 1-slice lookahead
// (TENSORcnt-tracked, one barrier per slice). A single z pass merges each
// slice's window max into all adaptive-D bins containing it (bins overlap by
// <=1 slice), avoiding the ~30% DMA re-reads an od-outer loop would cost.
// No matmul structure -> WMMA not applicable; TDM is the CDNA5 path here.

typedef __attribute__((ext_vector_type(4))) unsigned int v4u;
typedef __attribute__((ext_vector_type(8))) int v8i;
typedef __attribute__((ext_vector_type(4))) int v4i;

#define POOLED 8
#define CC 128
#define DD 32
#define HH 64
#define WW 64
#define SLICE (HH * WW)  // floats per LDS slice buffer

__device__ __forceinline__ int bin_start(int i, int L) { return (i * L) >> 3; }
__device__ __forceinline__ int bin_end(int i, int L) { return ((i + 1) * L + 7) >> 3; }

// Issue a TDM load of a 2D tile (Ly rows x Lx f32, row stride = WW elements)
// from global byte address `gaddr` into LDS byte offset `lds_off` (packed).
__device__ __forceinline__ void tdm_load_2d(uint64_t gaddr, uint32_t lds_off,
                                            int Lx, int Ly) {
  const uint32_t td0 = (uint32_t)Lx;        // tensor_dim0 (== tile: fully in-bounds)
  const uint32_t td1 = (uint32_t)Ly;        // tensor_dim1
  const uint32_t s0  = (uint32_t)WW;        // tensor_dim0_stride (elements)
  const uint32_t s1  = (uint32_t)(HH * WW); // tensor_dim1_stride (unused for 2D tile)

  // D# group 0 (128b): count=1 valid user desc, lds_addr, global_addr, type=2
  v4u g0;
  g0.x = 1u;
  g0.y = lds_off;
  g0.z = (uint32_t)gaddr;
  g0.w = ((uint32_t)(gaddr >> 32) & 0x01FFFFFFu) | (2u << 30);

  // D# group 1 (256b): wg_mask=0, data_size=2 (4B), no barrier/iterate/pad,
  // tensor dims, tile dims (tile_dim2=0 -> 2D), strides.
  v8i g1;
  g1[0] = (int)(2u << 16);                              // data_size=4B
  g1[1] = (int)((td0 & 0xFFFFu) << 16);                 // [63:48] = tensor_dim0 lo16
  g1[2] = (int)((td0 >> 16) | ((td1 & 0xFFFFu) << 16)); // td0 hi16 | td1 lo16
  g1[3] = (int)((td1 >> 16) | ((uint32_t)Lx << 16));    // td1 hi16 | tile_dim0
  g1[4] = (int)((uint32_t)Ly & 0xFFFFu);                // tile_dim1 | tile_dim2=0
  g1[5] = (int)s0;                                      // stride0[31:0]
  g1[6] = (int)((s1 & 0xFFFFu) << 16);                  // stride0[47:32]=0 | stride1 lo16
  g1[7] = (int)(s1 >> 16);                              // stride1[47:16]

  v4i g2 = {0, 0, 0, 0};
  v4i g3 = {0, 0, 0, 0};
  v8i g4 = {0, 0, 0, 0, 0, 0, 0, 0};  // extra operand on 6-arg toolchain
#if __has_builtin(__builtin_amdgcn_tensor_load_to_lds)
  __builtin_amdgcn_tensor_load_to_lds(g0, g1, g2, g3, g4, 0);
#else
  asm volatile("tensor_load_to_lds %0, %1, %2, %3"
               :
               : "s"(g0), "s"(g1), "s"(g2), "s"(g3)
               : "memory");
#endif
}

__global__ __launch_bounds__(64) void roi_pool3d_tdm(
    const float* __restrict__ feat, const float* __restrict__ rois,
    float* __restrict__ out) {
  __shared__ float tile[3][SLICE];  // 3 x 16 KB rotating slice buffers

  const int c = blockIdx.x;
  const int n = blockIdx.y;

  const float* r = rois + (size_t)n * 7;
  const int b    = (int)r[0];
  const int xmin = (int)r[1];
  const int ymin = (int)r[2];
  const int zmin = (int)r[3];
  const int Lx   = (int)r[4] - xmin;
  const int Ly   = (int)r[5] - ymin;
  const int Lz   = (int)r[6] - zmin;

  const int t  = threadIdx.x;
  const int oh = t >> 3;
  const int ow = t & 7;
  const int ys = bin_start(oh, Ly), ye = bin_end(oh, Ly);
  const int xs = bin_start(ow, Lx), xe = bin_end(ow, Lx);

  // Adaptive-D bin bounds (tile-local z). Adjacent bins overlap by <=1 slice.
  int zlo[POOLED], zhi[POOLED];
#pragma unroll
  for (int od = 0; od < POOLED; ++od) {
    zlo[od] = bin_start(od, Lz);
    zhi[od] = bin_end(od, Lz);
  }

  // Global byte address of element (b, c, z=0, ymin, xmin)
  const uint64_t base =
      (uint64_t)(uintptr_t)feat +
      4ull * ((uint64_t)(b * CC + c) * (DD * HH * WW) +
              (uint64_t)ymin * WW + (uint64_t)xmin);
  const uint32_t lds0 = (uint32_t)(uintptr_t)&tile[0][0];
  const uint64_t zstride = 4ull * (HH * WW);

  float acc[POOLED];
#pragma unroll
  for (int od = 0; od < POOLED; ++od) acc[od] = -__builtin_inff();

  // Wave 0 drives the TDM engine (wave-granular branch: TDM ignores EXEC).
  if (t < 32) {
    tdm_load_2d(base + (uint64_t)zmin * zstride, lds0, Lx, Ly);
  }

  int cur = 0, nxt = 1;
  for (int i = 0; i < Lz; ++i) {
    if (t < 32) {
      if (i + 1 < Lz) {
        // Prefetch slice i+1; buf[(i+1)%3] last held slice i-2, whose readers
        // all passed barrier(i-1) -> safe to overwrite. Then wait for slice i
        // (TDM ops retire in order: outstanding = {i, i+1}, so wait cnt<=1).
        tdm_load_2d(base + (uint64_t)(zmin + i + 1) * zstride,
                    lds0 + (uint32_t)(nxt * (SLICE * 4)), Lx, Ly);
        __builtin_amdgcn_s_wait_tensorcnt(1);
      } else {
        __builtin_amdgcn_s_wait_tensorcnt(0);
      }
    }
    __syncthreads();  // publish DMA'd slice i to both waves

    // Window max for this thread's (oh,ow) bin over slice i.
    const float* s = &tile[0][0] + cur * SLICE;
    float m = -__builtin_inff();
    for (int y = ys; y < ye; ++y) {
      const float* row = s + y * Lx;
      for (int x = xs; x < xe; ++x) m = fmaxf(m, row[x]);
    }

    // Merge into every adaptive-D bin containing slice i (at most 2).
#pragma unroll
    for (int od = 0; od < POOLED; ++od) {
      if (i >= zlo[od] && i < zhi[od]) acc[od] = fmaxf(acc[od], m);
    }

    cur = nxt;
    nxt = (nxt == 2) ? 0 : nxt + 1;
  }

  float* outp = out + ((((size_t)n * CC + c) * POOLED) << 6);
#pragma unroll
  for (int od = 0; od < POOLED; ++od) {
    outp[(od << 6) + t] = acc[od];  // 64 consecutive floats, coalesced
  }
}

extern "C" void kernel_launch(void* const* d_in, const int* in_sizes, int n_in,
                              void* d_out, int out_size, void* d_ws, size_t ws_size,
                              hipStream_t stream) {
  const float* feat = (const float*)d_in[0];  // [2,128,32,64,64] f32
  const float* rois = (const float*)d_in[1];  // [N,7] f32
  float* out = (float*)d_out;                 // [N,128,8,8,8] f32
  const int N = in_sizes[1] / 7;

  dim3 grid(CC, N);
  roi_pool3d_tdm<<<grid, 64, 0, stream>>>(feat, rois, out);
}